// DifferentiableMemory_24601572671490
// MI455X (gfx1250) — compile-verified
//
#include <hip/hip_runtime.h>

typedef __attribute__((ext_vector_type(2))) float v2f;
typedef __attribute__((ext_vector_type(4))) float v4f;
typedef __attribute__((ext_vector_type(8))) float v8f;

#define BQ    4096
#define NK    50000
#define DD    128
#define KT    50
#define EPSW  1e-7f
#define QT    16          // queries per block (= WMMA M)
#define CT    16          // columns per WMMA tile (= WMMA N)
#define NT    4           // column tiles per wave-iteration (4 indep. WMMA chains)
#define PD    4           // software-pipeline depth (K-steps of B prefetch)
#define WAVES 4
#define SEG   4           // N split across blockIdx.y for grid parallelism
#define BLOCK (WAVES * 32)
#define FMAX  3.402823466e+38f

// ---------------------------------------------------------------------------
// Kernel 1: m2[n] = sum_d mem_keys[n][d]^2
// ---------------------------------------------------------------------------
__global__ void dm_m2_kernel(const float* __restrict__ keys, float* __restrict__ m2) {
  int n = blockIdx.x * blockDim.x + threadIdx.x;
  if (n >= NK) return;
  const v4f* kp = (const v4f*)(keys + (size_t)n * DD);
  float acc = 0.f;
#pragma unroll
  for (int i = 0; i < DD / 4; ++i) {
    v4f v = kp[i];
    acc += v.x * v.x + v.y * v.y + v.z * v.z + v.w * v.w;
  }
  m2[n] = acc;
}

// ---------------------------------------------------------------------------
// Kernel 2: fused fp32-WMMA GEMM + per-row top-50 over one N-segment.
//   score[m][n] = m2[n] - 2 * (q[m] . k[n])   (rank == rank of squared L2)
// Output: per-(row, segment) ascending top-50 (dist, idx) lists in workspace.
// A fragment register-resident (64 VGPRs); 4 independent WMMA chains;
// B software-pipelined PD K-steps ahead (modulo buffer, ~16 loads in flight).
// ---------------------------------------------------------------------------
__global__ __launch_bounds__(BLOCK)
__attribute__((amdgpu_waves_per_eu(1)))
void dm_topk_kernel(
    const float* __restrict__ q, const float* __restrict__ keys,
    const float* __restrict__ m2,
    float* __restrict__ gdist, int* __restrict__ gidx) {
  __shared__ float sQ[QT][DD];                   // 8 KB   query tile
  __shared__ float sScore[WAVES][QT][CT * NT];   // 16 KB  per-wave 16x64 scores
  __shared__ float sDist[WAVES][QT][KT];         // 12.8 KB per-wave top-K dists
  __shared__ int   sIdx[WAVES][QT][KT];          // 12.8 KB per-wave top-K idx

  const int tid   = threadIdx.x;
  const int wave  = tid >> 5;
  const int lane  = tid & 31;
  const int l16   = lane & 15;
  const int half  = lane >> 4;                   // 0: K pair {k,k+1}, 1: {k+2,k+3}
  const int qbase = blockIdx.x * QT;
  const int seg   = blockIdx.y;

  // Stage query tile (coalesced) and init candidate lists.
  for (int i = tid; i < QT * DD; i += BLOCK)
    sQ[i >> 7][i & (DD - 1)] = q[(size_t)qbase * DD + i];
  for (int i = tid; i < WAVES * QT * KT; i += BLOCK) {
    (&sDist[0][0][0])[i] = FMAX;
    (&sIdx[0][0][0])[i]  = 0;
  }
  __syncthreads();

  // Hoist the A fragment (this lane's K pairs of query row l16) into VGPRs.
  v2f aK[DD / 4];
#pragma unroll
  for (int i = 0; i < DD / 4; ++i) {
    const int k0 = 4 * i + 2 * half;
    aK[i].x = sQ[l16][k0];
    aK[i].y = sQ[l16][k0 + 1];
  }

  float curmax = FMAX;   // K-th threshold for the row owned by lane (<16)
  int   argmax = 0;

  const int NTILES   = NK / CT;                      // 3125 (exact)
  const int NST      = (NTILES + NT - 1) / NT;       // 782 supertiles (64 cols)
  const int NSTRIPES = WAVES * SEG;                  // 16
  const int stripe   = wave + WAVES * seg;
  const int ITERS    = (NST + NSTRIPES - 1) / NSTRIPES;  // 49, uniform

  // B viewed as array of v2f (8B) off a uniform SGPR base -> 32-bit voffsets.
  const v2f* kb = (const v2f*)keys;

  for (int it = 0; it < ITERS; ++it) {
    const int st   = stripe + NSTRIPES * it;         // this wave's supertile
    const int col0 = st * (CT * NT);

    // Per-tile 32-bit v2f-element base indices, clamped in-bounds (uniform
    // control flow; the selection scan masks out-of-range columns via ncols).
    int bi[NT];
#pragma unroll
    for (int j = 0; j < NT; ++j) {
      int tj = st * NT + j;
      if (tj > NTILES - 1) tj = NTILES - 1;
      bi[j] = tj * (CT * DD / 2) + l16 * (DD / 2) + half;
    }

    // Software pipeline: preload PD K-steps of B into a modulo buffer.
    v2f bb[PD][NT];
#pragma unroll
    for (int p = 0; p < PD; ++p)
#pragma unroll
      for (int j = 0; j < NT; ++j)
        bb[p][j] = kb[bi[j] + 2 * p];

    // 4 independent 16x16 accumulator chains; A operand reused 4x per step.
    v8f c[NT] = {};
#pragma unroll
    for (int s = 0; s < DD / 4; ++s) {
      const v2f a = aK[s];
#pragma unroll
      for (int j = 0; j < NT; ++j)
        c[j] = __builtin_amdgcn_wmma_f32_16x16x4_f32(
                   false, a, false, bb[s & (PD - 1)][j], (short)0, c[j],
                   false, false);
      if (s + PD < DD / 4) {                         // refill consumed slot
#pragma unroll
        for (int j = 0; j < NT; ++j)
          bb[s & (PD - 1)][j] = kb[bi[j] + 2 * (s + PD)];
      }
    }

    // Scores -> LDS. C/D layout: lane holds col N=l16, rows M=r+8*half.
#pragma unroll
    for (int j = 0; j < NT; ++j) {
      int cj = col0 + CT * j + l16;
      if (cj > NK - 1) cj = NK - 1;
      const float m2v = m2[cj];
#pragma unroll
      for (int r = 0; r < 8; ++r)
        sScore[wave][r + 8 * half][CT * j + l16] = m2v - 2.0f * c[j][r];
    }
    __syncthreads();

    // Lanes 0..15 each own one query row: threshold-test up to 64 new scores;
    // rare O(K) rescan on actual insertion (~E[210] total per row per stripe).
    const int ncols = (NK - col0) < (CT * NT) ? (NK - col0) : (CT * NT);
    if (lane < QT && ncols > 0) {
      const int m = lane;
      for (int n = 0; n < ncols; ++n) {
        const float s = sScore[wave][m][n];
        if (s < curmax) {
          sDist[wave][m][argmax] = s;
          sIdx[wave][m][argmax]  = col0 + n;
          float mx = -FMAX; int ai = 0;
          for (int jj = 0; jj < KT; ++jj) {
            const float dj = sDist[wave][m][jj];
            if (dj > mx) { mx = dj; ai = jj; }
          }
          curmax = mx; argmax = ai;
        }
      }
    }
    __syncthreads();
  }

  // Merge this block's WAVES lists per row; emit ascending top-50 to global.
  if (wave == 0 && lane < QT) {
    const int    m    = lane;
    const size_t base = ((size_t)(qbase + m) * SEG + seg) * KT;
    for (int kk = 0; kk < KT; ++kk) {
      float best = FMAX; int bw = 0, bj = 0;
      for (int w = 0; w < WAVES; ++w)
        for (int j = 0; j < KT; ++j) {
          const float dv = sDist[w][m][j];
          if (dv < best) { best = dv; bw = w; bj = j; }
        }
      sDist[bw][m][bj] = FMAX;                 // mark consumed
      gdist[base + kk] = best;                 // ascending order
      gidx[base + kk]  = sIdx[bw][m][bj];
    }
  }
}

// ---------------------------------------------------------------------------
// Kernel 3: per-row 4-way merge of sorted segment lists + weighting.
// ---------------------------------------------------------------------------
__global__ void dm_finalize_kernel(
    const float* __restrict__ q, const float* __restrict__ vals,
    const float* __restrict__ gdist, const int* __restrict__ gidx,
    float* __restrict__ out) {
  int r = blockIdx.x * blockDim.x + threadIdx.x;
  if (r >= BQ) return;

  const v4f* qp = (const v4f*)(q + (size_t)r * DD);
  float q2 = 0.f;
#pragma unroll
  for (int i = 0; i < DD / 4; ++i) {
    v4f v = qp[i];
    q2 += v.x * v.x + v.y * v.y + v.z * v.z + v.w * v.w;
  }

  int cur[SEG] = {0, 0, 0, 0};
  float wsum = 0.f, wv = 0.f;
  for (int k = 0; k < KT; ++k) {
    float best = FMAX; int bs = 0;
#pragma unroll
    for (int s = 0; s < SEG; ++s) {
      if (cur[s] < KT) {
        const float dv = gdist[((size_t)r * SEG + s) * KT + cur[s]];
        if (dv < best) { best = dv; bs = s; }
      }
    }
    const int id = gidx[((size_t)r * SEG + bs) * KT + cur[bs]];
    cur[bs]++;
    const float d2  = best + q2;               // true squared L2
    const float wgt = 1.0f / (d2 + EPSW);
    wsum += wgt;
    wv   += wgt * vals[id];
  }
  out[r] = wv / wsum;
}

// ---------------------------------------------------------------------------
extern "C" void kernel_launch(void* const* d_in, const int* in_sizes, int n_in,
                              void* d_out, int out_size, void* d_ws, size_t ws_size,
                              hipStream_t stream) {
  (void)in_sizes; (void)n_in; (void)out_size; (void)ws_size;
  const float* q    = (const float*)d_in[0];   // [B, D]
  const float* keys = (const float*)d_in[1];   // [N, D]
  const float* vals = (const float*)d_in[2];   // [N]
  float* out = (float*)d_out;                  // [B]

  // Workspace layout (floats): m2[50176 pad] | gdist[BQ*SEG*KT] | gidx[...]
  float* wsf   = (float*)d_ws;
  float* m2    = wsf;                              // 50000 (+pad)
  float* gdist = wsf + 50176;                      // 819200
  int*   gidx  = (int*)(wsf + 50176 + (size_t)BQ * SEG * KT);

  dm_m2_kernel<<<(NK + 255) / 256, 256, 0, stream>>>(keys, m2);
  dim3 grid(BQ / QT, SEG);
  dm_topk_kernel<<<grid, BLOCK, 0, stream>>>(q, keys, m2, gdist, gidx);
  dm_finalize_kernel<<<(BQ + 127) / 128, 128, 0, stream>>>(q, vals, gdist, gidx, out);
}